// FastRCNN_42451456753989
// MI455X (gfx1250) — compile-verified
//
#include <hip/hip_runtime.h>
#include <hip/hip_bf16.h>
#include <math.h>

// ---------------------------------------------------------------------------
// Fast-RCNN head for MI455X (gfx1250, wave32).
//   roi_pool(2x2 max)+mean  -> roi_feat (2048,1280)           [ws]
//   h1 = relu(roi_feat@W1+b1)   V_WMMA_F32_16X16X4_F32, MT=4  [ws]
//   h2 = h1@W2+b2               V_WMMA_F32_16X16X4_F32, MT=4  [ws]
//   [cls|det] = h2@[Wcls|Wdet|0] + [bcls|bdet|0]  (padded N=32 WMMA GEMM)
//   max_iou = rowmax(batch-masked IoU(proposals, gt))
// All GEMM dims are compile-time constants so A and B addresses fold into
// the 24-bit immediate offsets of global_load -- no per-iteration 64-bit
// VALU address arithmetic in the WMMA loop.
// ---------------------------------------------------------------------------

typedef __attribute__((ext_vector_type(2))) float v2f;
typedef __attribute__((ext_vector_type(8))) float v8f;

// Problem constants (fixed by the reference setup_inputs()).
#define NB   8
#define NC   1280
#define NH   7
#define NW   7
#define NM   2048     // proposals
#define NG   128      // gt boxes
#define HID  256
#define NCLS 21       // ncls+1
#define NHEAD 25      // 21 cls + 4 det
#define NPAD 32       // padded head width for WMMA

// ---------------------------------------------------------------------------
// Kernel 1: roi_pool (2x2, spatial_scale=1) + mean over bins.
// One block per proposal; each thread walks NC/blockDim channels.
// Bin bounds are integer-valued after round/floor/ceil/clip, so the reference
// masked-max is exactly a max over the integer window; empty bin -> 0.
// ---------------------------------------------------------------------------
__global__ __launch_bounds__(256) void roi_pool_kernel(
    const float* __restrict__ feat,       // (NB,NC,NH,NW)
    const float* __restrict__ proposals,  // (NM,4)
    const int*   __restrict__ pbids,      // (NM)
    float*       __restrict__ roi_feat)   // (NM,NC)
{
    const int m = blockIdx.x;

    // rintf == round-half-to-even, matching jnp.round
    const float x1 = rintf(proposals[m * 4 + 0]);
    const float y1 = rintf(proposals[m * 4 + 1]);
    const float x2 = rintf(proposals[m * 4 + 2]);
    const float y2 = rintf(proposals[m * 4 + 3]);
    const float roi_w = fmaxf(x2 - x1 + 1.0f, 1.0f);
    const float roi_h = fmaxf(y2 - y1 + 1.0f, 1.0f);
    const float bin_w = roi_w * 0.5f;
    const float bin_h = roi_h * 0.5f;

    int ys[2], ye[2], xs[2], xe[2];
#pragma unroll
    for (int i = 0; i < 2; ++i) {
        float hs = fminf(fmaxf(floorf((float)i * bin_h) + y1, 0.0f), (float)NH);
        float he = fminf(fmaxf(ceilf((float)(i + 1) * bin_h) + y1, 0.0f), (float)NH);
        float ws = fminf(fmaxf(floorf((float)i * bin_w) + x1, 0.0f), (float)NW);
        float we = fminf(fmaxf(ceilf((float)(i + 1) * bin_w) + x1, 0.0f), (float)NW);
        ys[i] = (int)hs; ye[i] = (int)he;   // integer-valued floats
        xs[i] = (int)ws; xe[i] = (int)we;
    }

    const int b = pbids[m];
    const float* __restrict__ fb = feat + (size_t)b * NC * NH * NW;

    for (int c = threadIdx.x; c < NC; c += blockDim.x) {
        const float* __restrict__ fc = fb + (size_t)c * (NH * NW);
        float sum = 0.0f;
#pragma unroll
        for (int i = 0; i < 2; ++i) {
#pragma unroll
            for (int j = 0; j < 2; ++j) {
                float mx = 0.0f;                       // empty bin -> 0
                if (ys[i] < ye[i] && xs[j] < xe[j]) {
                    mx = -INFINITY;
                    for (int y = ys[i]; y < ye[i]; ++y)
                        for (int x = xs[j]; x < xe[j]; ++x)
                            mx = fmaxf(mx, fc[y * NW + x]);
                }
                sum += mx;
            }
        }
        roi_feat[(size_t)m * NC + c] = sum * 0.25f;    // mean over 2x2 bins
    }
}

// ---------------------------------------------------------------------------
// f32 GEMM  C = op(A(MxK) @ B(KxN) + bias)  via V_WMMA_F32_16X16X4_F32.
// One wave computes a (16*MT) x 16 strip of C: MT accumulators share one B
// fragment per k-step (4x fewer strided scalar B loads per WMMA). K *and* N
// are template constants so all load/store addresses fold into immediate
// offsets (max B offset = 1280*256*4 = 1.3 MB < 24-bit IOFFSET range).
// Layouts per ISA 7.12.2:
//   A 16x4 : lanes 0-15 row M=l, {K=0,K=1} in v2f; lanes 16-31 {K=2,K=3}
//   B 4x16 : lanes 0-15 rows K=0..1 (one per VGPR), lanes 16-31 rows K=2..3
//   C/D    : acc[v] = row (v + 8*laneHalf), col = lane&15
// Tile counts divide the launch exactly, so EXEC stays all-ones.
// ---------------------------------------------------------------------------
template <int KSIZE, int NSIZE, int MT, bool RELU>
__global__ __launch_bounds__(256) void gemm_wmma_f32_kernel(
    const float* __restrict__ A,     // (M,KSIZE) row-major
    const float* __restrict__ B,     // (KSIZE,NSIZE) row-major
    const float* __restrict__ bias,  // (NSIZE)
    float*       __restrict__ C)     // (M,NSIZE) row-major
{
    const int lane = threadIdx.x & 31;
    const int wave = blockIdx.x * 8 + (threadIdx.x >> 5);
    constexpr int NTN = NSIZE / 16;
    const int tm   = wave / NTN;                 // wave-uniform
    const int tn   = wave - tm * NTN;

    const int half = lane >> 4;                  // 0: K pair {0,1}, 1: {2,3}
    const int l16  = lane & 15;

    const float* __restrict__ a0 = A + (size_t)(tm * (16 * MT) + l16) * KSIZE + 2 * half;
    const float* __restrict__ bp = B + (size_t)(2 * half) * NSIZE + tn * 16 + l16;

    v8f acc[MT];
#pragma unroll
    for (int t = 0; t < MT; ++t)
#pragma unroll
        for (int e = 0; e < 8; ++e) acc[t][e] = 0.0f;

#pragma unroll 8
    for (int k0 = 0; k0 < KSIZE; k0 += 4) {
        v2f b;
        b.x = bp[k0 * NSIZE];                    // B[k0+2h  ][n]  (imm offset)
        b.y = bp[(k0 + 1) * NSIZE];              // B[k0+2h+1][n]  (imm offset)
#pragma unroll
        for (int t = 0; t < MT; ++t) {
            v2f a = *(const v2f*)(a0 + t * 16 * KSIZE + k0);   // imm offset
            acc[t] = __builtin_amdgcn_wmma_f32_16x16x4_f32(
                /*neg_a=*/false, a, /*neg_b=*/false, b,
                /*c_mod=*/(short)0, acc[t], /*reuse_a=*/false, /*reuse_b=*/false);
        }
    }

    const int   col = tn * 16 + l16;
    const float bv  = bias[col];
    float* __restrict__ cp = C + (size_t)(tm * (16 * MT) + 8 * half) * NSIZE + col;
#pragma unroll
    for (int t = 0; t < MT; ++t) {
#pragma unroll
        for (int v = 0; v < 8; ++v) {
            float val = acc[t][v] + bv;
            if (RELU) val = fmaxf(val, 0.0f);
            cp[(t * 16 + v) * NSIZE] = val;      // imm offsets
        }
    }
}

// ---------------------------------------------------------------------------
// Pack [Wcls | Wdet | zeros] -> Wp (HID x 32), [bcls | bdet | zeros] -> biasp.
// ---------------------------------------------------------------------------
__global__ __launch_bounds__(256) void pack_heads_kernel(
    const float* __restrict__ Wcls, const float* __restrict__ bcls,
    const float* __restrict__ Wdet, const float* __restrict__ bdet,
    float* __restrict__ Wp, float* __restrict__ biasp)
{
    const int idx = blockIdx.x * blockDim.x + threadIdx.x;
    if (idx < HID * NPAD) {
        const int k = idx >> 5;
        const int n = idx & (NPAD - 1);
        float v = 0.0f;
        if (n < NCLS)            v = Wcls[(size_t)k * NCLS + n];
        else if (n < NHEAD)      v = Wdet[(size_t)k * 4 + (n - NCLS)];
        Wp[idx] = v;
    } else if (idx < HID * NPAD + NPAD) {
        const int n = idx - HID * NPAD;
        float v = 0.0f;
        if (n < NCLS)            v = bcls[n];
        else if (n < NHEAD)      v = bdet[n - NCLS];
        biasp[n] = v;
    }
}

// ---------------------------------------------------------------------------
// Scatter padded head output (NM x 32) into cls (NM x 21) and det (NM x 4).
// ---------------------------------------------------------------------------
__global__ __launch_bounds__(256) void scatter_heads_kernel(
    const float* __restrict__ h3,      // (NM, NPAD)
    float* __restrict__ out_cls,       // (NM, NCLS)
    float* __restrict__ out_det)       // (NM, 4)
{
    const int idx = blockIdx.x * blockDim.x + threadIdx.x;
    if (idx >= NM * NHEAD) return;
    const int m = idx / NHEAD;
    const int n = idx - m * NHEAD;
    const float v = h3[(size_t)m * NPAD + n];
    if (n < NCLS) out_cls[(size_t)m * NCLS + n] = v;
    else          out_det[(size_t)m * 4 + (n - NCLS)] = v;
}

// ---------------------------------------------------------------------------
// Batch-masked pairwise IoU row-max. Uses RAW (unrounded) proposals,
// matching the reference. Masked entries contribute 0, so init best = 0.
// ---------------------------------------------------------------------------
__global__ __launch_bounds__(256) void iou_kernel(
    const float* __restrict__ proposals,  // (NM,4)
    const float* __restrict__ bboxes,     // (NG,5)
    const int*   __restrict__ pbids,      // (NM)
    const int*   __restrict__ gbids,      // (NG)
    float*       __restrict__ max_iou)    // (NM)
{
    const int m = blockIdx.x * blockDim.x + threadIdx.x;
    if (m >= NM) return;
    const float px1 = proposals[m * 4 + 0];
    const float py1 = proposals[m * 4 + 1];
    const float px2 = proposals[m * 4 + 2];
    const float py2 = proposals[m * 4 + 3];
    const float ap  = (px2 - px1) * (py2 - py1);
    const int   pb  = pbids[m];

    float best = 0.0f;
    for (int n = 0; n < NG; ++n) {
        if (gbids[n] != pb) continue;
        const float gx1 = bboxes[n * 5 + 0];
        const float gy1 = bboxes[n * 5 + 1];
        const float gx2 = bboxes[n * 5 + 2];
        const float gy2 = bboxes[n * 5 + 3];
        float ix = fmaxf(fminf(px2, gx2) - fmaxf(px1, gx1), 0.0f);
        float iy = fmaxf(fminf(py2, gy2) - fmaxf(py1, gy1), 0.0f);
        float inter = ix * iy;
        float ag    = (gx2 - gx1) * (gy2 - gy1);
        float uni   = fmaxf(ap + ag - inter, 1e-6f);
        best = fmaxf(best, inter / uni);
    }
    max_iou[m] = best;
}

// ---------------------------------------------------------------------------
extern "C" void kernel_launch(void* const* d_in, const int* in_sizes, int n_in,
                              void* d_out, int out_size, void* d_ws, size_t ws_size,
                              hipStream_t stream)
{
    const float* feature_map = (const float*)d_in[0];   // (8,1280,7,7)
    const float* proposals   = (const float*)d_in[1];   // (2048,4)
    const float* bboxes      = (const float*)d_in[2];   // (128,5)
    const float* W1          = (const float*)d_in[3];   // (1280,256)
    const float* b1          = (const float*)d_in[4];   // (256)
    const float* W2          = (const float*)d_in[5];   // (256,256)
    const float* b2          = (const float*)d_in[6];   // (256)
    const float* Wcls        = (const float*)d_in[7];   // (256,21)
    const float* bcls        = (const float*)d_in[8];   // (21)
    const float* Wdet        = (const float*)d_in[9];   // (256,4)
    const float* bdet        = (const float*)d_in[10];  // (4)
    const int*   pbids       = (const int*)d_in[11];    // (2048)
    const int*   gbids       = (const int*)d_in[12];    // (128)

    float* out     = (float*)d_out;
    float* out_cls = out;                       // (2048,21)
    float* out_det = out + (size_t)NM * NCLS;   // (2048,4)
    float* out_iou = out_det + (size_t)NM * 4;  // (2048)

    char* ws = (char*)d_ws;
    float* roi_feat = (float*)(ws);                      // 10,485,760 B
    float* h1       = (float*)(ws + 10485760u);          //  2,097,152 B
    float* h2       = (float*)(ws + 12582912u);          //  2,097,152 B
    float* h3       = (float*)(ws + 14680064u);          //    262,144 B (2048x32)
    float* Wp       = (float*)(ws + 14942208u);          //     32,768 B (256x32)
    float* biasp    = (float*)(ws + 14974976u);          //        128 B (32)

    // 0) pack padded head weights (independent of everything else)
    pack_heads_kernel<<<(HID * NPAD + NPAD + 255) / 256, 256, 0, stream>>>(
        Wcls, bcls, Wdet, bdet, Wp, biasp);

    // 1) ROI pool + mean
    roi_pool_kernel<<<NM, 256, 0, stream>>>(feature_map, proposals, pbids, roi_feat);

    // 2) h1 = relu(roi_feat @ W1 + b1): (2048x1280)@(1280x256)
    //    waves = (2048/64)*(256/16) = 512 -> 64 blocks of 8 waves
    gemm_wmma_f32_kernel<NC, HID, 4, true><<<64, 256, 0, stream>>>(
        roi_feat, W1, b1, h1);

    // 3) h2 = h1 @ W2 + b2: (2048x256)@(256x256); 512 waves -> 64 blocks
    gemm_wmma_f32_kernel<HID, HID, 4, false><<<64, 256, 0, stream>>>(
        h1, W2, b2, h2);

    // 4) heads: (2048x256)@(256x32 padded); waves = 32*2 = 64 -> 8 blocks
    gemm_wmma_f32_kernel<HID, NPAD, 4, false><<<8, 256, 0, stream>>>(
        h2, Wp, biasp, h3);

    // 4b) scatter padded head output into cls/det layouts
    scatter_heads_kernel<<<(NM * NHEAD + 255) / 256, 256, 0, stream>>>(
        h3, out_cls, out_det);

    // 5) masked IoU row-max
    iou_kernel<<<(NM + 255) / 256, 256, 0, stream>>>(
        proposals, bboxes, pbids, gbids, out_iou);
}